// interactionModule_44667659879080
// MI455X (gfx1250) — compile-verified
//
#include <hip/hip_runtime.h>
#include <hip/hip_bf16.h>

#define INV_TWO_PI 0.15915494309189535f
#define K1_CUTOFF  0.13986588f   // _k1(2.0) via the x<=2 polynomial branch, f32

// ---------------------------------------------------------------------------
// Kernel 1: zero the per-node accumulators [N][4] = {sum_vx, sum_vy, sum_w, deg}
// ---------------------------------------------------------------------------
__global__ void im_zero_kernel(float* __restrict__ acc, int n4) {
    int i = blockIdx.x * blockDim.x + threadIdx.x;
    if (i < n4) acc[i] = 0.0f;
}

// ---------------------------------------------------------------------------
// Kernel 2: per-edge messages, scatter-add into acc[dst][0..3]
// ---------------------------------------------------------------------------
__global__ __launch_bounds__(256) void im_edge_kernel(
    const float* __restrict__ x,        // [N,2]
    const float* __restrict__ theta,    // [N,1]
    const float* __restrict__ kappa_p,
    const float* __restrict__ r_p,
    const float* __restrict__ beta_p,
    const float* __restrict__ A_CIL_p,
    const float* __restrict__ A_CFs,    // [2]
    const float* __restrict__ A_chems,  // [2]
    const int*   __restrict__ celltype, // [N]
    const int*   __restrict__ src,      // [E]
    const int*   __restrict__ dst,      // [E]
    float*       __restrict__ acc,      // [N,4]
    int E)
{
    int i = blockIdx.x * blockDim.x + threadIdx.x;
    if (i >= E) return;

    // All uniform parameters: these become scalar-cache s_loads, no per-edge
    // vector gathers. The 2-entry embedding tables are fully preloaded and
    // selected with v_cndmask instead of a dependent VGPR-indexed load.
    const float kappa   = kappa_p[0];
    const float r       = r_p[0];
    const float beta    = beta_p[0];
    const float A_CIL   = A_CIL_p[0];
    const float A_CF0   = A_CFs[0],   A_CF1   = A_CFs[1];
    const float A_chem0 = A_chems[0], A_chem1 = A_chems[1];

    // Edge index streams are read exactly once: non-temporal loads keep the
    // reused node data (x/theta/celltype/acc) resident in WGP$/L2.
    int s = __builtin_nontemporal_load(src + i);
    int d = __builtin_nontemporal_load(dst + i);

    float2 xs = *(const float2*)(x + 2 * (size_t)s);
    float2 xd = *(const float2*)(x + 2 * (size_t)d);
    float  th_s = theta[s];
    float  th_d = theta[d];
    int    ct   = celltype[d];

    float drx = xs.x - xd.x;
    float dry = xs.y - xd.y;
    float abs_dr = sqrtf(drx * drx + dry * dry);
    float inv_dr = 1.0f / fmaxf(abs_dr, 1e-12f);
    float ux = drx * inv_dr;
    float uy = dry * inv_dr;

    float sn_s, cs_s, sn_d, cs_d;
    sincosf(th_s, &sn_s, &cs_s);   // p_nb = (cos, sin)(theta[src])
    sincosf(th_d, &sn_d, &cs_d);   // p_tg = (cos, sin)(theta[dst])

    float drp_inner = ux * cs_s + uy * sn_s;
    float drp_cross = ux * sn_d - uy * cs_d;

    float J_CIL = r / abs_dr - 1.0f;
    float J_CF  = 0.5f * (1.0f + drp_inner);

    // Modified Bessel K1 (A&S 9.8.7 / 9.8.8), matching the reference.
    float xk = kappa * abs_dr;
    float k1v;
    if (xk <= 2.0f) {
        // Rare branch (dist < 2 in a 100x100 box): <0.2% of edges, so most
        // waves skip the logf path entirely.
        float q = xk / 3.75f;
        float t = q * q;
        float i1 = xk * (0.5f + t * (0.87890594f + t * (0.51498869f
                 + t * (0.15084934f + t * (0.02658733f + t * (0.00301532f
                 + t * 0.00032411f))))));
        float y = xk * xk * 0.25f;
        float p = 1.0f + y * (0.15443144f + y * (-0.67278579f
                + y * (-0.18156897f + y * (-0.01919402f + y * (-0.00110404f
                + y * (-0.00004686f))))));
        k1v = logf(xk * 0.5f) * i1 + p / xk;
    } else {
        float z = 2.0f / xk;
        float p = 1.25331414f + z * (0.23498619f + z * (-0.03655620f
                + z * (0.01504268f + z * (-0.00780353f + z * (0.00325614f
                + z * (-0.00068245f))))));
        k1v = expf(-xk) * p / sqrtf(xk);
    }
    float J_chem = fmaxf(k1v - K1_CUTOFF, 0.0f) * (kappa * INV_TWO_PI);

    float A_CF   = ct ? A_CF1   : A_CF0;    // v_cndmask, no memory access
    float A_chem = ct ? A_chem1 : A_chem0;

    float mvx = -beta * J_CIL * ux;
    float mvy = -beta * J_CIL * uy;
    float mt  = (A_CF * J_CF - A_CIL * J_CIL) * drp_cross
              + A_chem * J_chem * drp_inner;

    // 4 no-return hardware f32 atomics, all into one 16B slot of the dst node
    // (same cacheline -> single L2 channel, fire-and-forget via STOREcnt).
    float* base = acc + 4 * (size_t)d;
    unsafeAtomicAdd(base + 0, mvx);
    unsafeAtomicAdd(base + 1, mvy);
    unsafeAtomicAdd(base + 2, mt);
    unsafeAtomicAdd(base + 3, 1.0f);
}

// ---------------------------------------------------------------------------
// Kernel 3: per-node finalize: mean mailbox + self-polarity terms
// ---------------------------------------------------------------------------
__global__ __launch_bounds__(256) void im_node_kernel(
    const float* __restrict__ theta,
    const float* __restrict__ u0_p,
    const float* __restrict__ A_ext_p,
    const float* __restrict__ acc,     // [N,4]
    float*       __restrict__ out_v,   // [N,2]
    float*       __restrict__ out_w,   // [N,1]
    int N)
{
    int i = blockIdx.x * blockDim.x + threadIdx.x;
    if (i >= N) return;

    const float u0    = u0_p[0];
    const float A_ext = A_ext_p[0];

    float4 a = *(const float4*)(acc + 4 * (size_t)i);
    float inv = 1.0f / fmaxf(a.w, 1.0f);

    float sn, cs;
    sincosf(theta[i], &sn, &cs);

    // Final outputs are write-once: non-temporal stores avoid displacing the
    // (tiny, hot) node-data lines.
    __builtin_nontemporal_store(a.x * inv + u0 * cs, out_v + 2 * (size_t)i + 0);
    __builtin_nontemporal_store(a.y * inv + u0 * sn, out_v + 2 * (size_t)i + 1);
    __builtin_nontemporal_store(a.z * inv + A_ext * cs, out_w + i);
}

// ---------------------------------------------------------------------------
extern "C" void kernel_launch(void* const* d_in, const int* in_sizes, int n_in,
                              void* d_out, int out_size, void* d_ws, size_t ws_size,
                              hipStream_t stream) {
    const float* x        = (const float*)d_in[0];
    const float* theta    = (const float*)d_in[1];
    const float* kappa    = (const float*)d_in[2];
    const float* r        = (const float*)d_in[3];
    const float* u0       = (const float*)d_in[4];
    const float* beta     = (const float*)d_in[5];
    const float* A_CIL    = (const float*)d_in[6];
    const float* A_ext    = (const float*)d_in[7];
    const float* A_CFs    = (const float*)d_in[8];
    const float* A_chems  = (const float*)d_in[9];
    const int*   celltype = (const int*)d_in[10];
    const int*   src      = (const int*)d_in[11];
    const int*   dst      = (const int*)d_in[12];

    const int N = in_sizes[0] / 2;
    const int E = in_sizes[11];

    float* acc   = (float*)d_ws;           // [N][4] accumulators, 16*N bytes
    float* out_v = (float*)d_out;          // [N,2] flat
    float* out_w = out_v + 2 * (size_t)N;  // [N,1] flat

    const int n4 = 4 * N;
    im_zero_kernel<<<(n4 + 255) / 256, 256, 0, stream>>>(acc, n4);

    im_edge_kernel<<<(E + 255) / 256, 256, 0, stream>>>(
        x, theta, kappa, r, beta, A_CIL, A_CFs, A_chems,
        celltype, src, dst, acc, E);

    im_node_kernel<<<(N + 255) / 256, 256, 0, stream>>>(
        theta, u0, A_ext, acc, out_v, out_w, N);
}